// RGCNClassifier_69999376990324
// MI455X (gfx1250) — compile-verified
//
#include <hip/hip_runtime.h>
#include <hip/hip_bf16.h>

// ---------------------------------------------------------------------------
// R-GCN classifier for MI455X (gfx1250, wave32, WMMA fp32 16x16x4)
//
// out[v] = relu( sum_r W_r @ (sum_{e: etype=r, dst=v} h[src(e)]) + W_self@h[v] + b )
//
// Fast path (if ws allows ~336 MB):
//   one edge pass -> agg8[8][N][128]; then ONE fused GEMM per layer that
//   accumulates all 8 relation tiles + the self tile in registers (bias in
//   init, ReLU at store) -- C is written exactly once per layer.
// Fallback path: per-relation scatter + beta=1 WMMA GEMMs.
// ---------------------------------------------------------------------------

typedef __attribute__((ext_vector_type(2))) float v2f;
typedef __attribute__((ext_vector_type(8))) float v8f;

#define N_NODES   65536
#define N_EDGES   786432
#define N_RELS    8
#define DIM       128
#define N_GRAPHS  64
#define N_CLASSES 10

#define LDA 132   // padded LDS stride: 132 % 64 == 4 -> conflict-free 16x4 frag loads
#define LDB 132

// ---- shared microkernel pieces -------------------------------------------

__device__ __forceinline__ void load_tiles_lds(float* sA, float* sB,
                                               const float* __restrict__ A,
                                               const float* __restrict__ B,
                                               int tid) {
  const float4* Ag = (const float4*)A;
  const float4* Bg = (const float4*)B;
#pragma unroll
  for (int i = 0; i < 16; ++i) {
    int idx = tid + i * 256;              // 0..4095 float4s
    float4 va = Ag[idx];
    int row = idx >> 5;                   // 32 float4 per row
    int k4  = (idx & 31) * 4;
    float* da = sA + row * LDA + k4;
    da[0] = va.x; da[1] = va.y; da[2] = va.z; da[3] = va.w;

    float4 vb = Bg[idx];
    int k  = idx >> 5;                    // B row = k
    int c4 = (idx & 31) * 4;              // B col base
    sB[(c4 + 0) * LDB + k] = vb.x;
    sB[(c4 + 1) * LDB + k] = vb.y;
    sB[(c4 + 2) * LDB + k] = vb.z;
    sB[(c4 + 3) * LDB + k] = vb.w;
  }
}

__device__ __forceinline__ void kloop_wmma(const float* sA, const float* sB,
                                           v8f acc[8], int m0, int lo, int hi) {
#pragma unroll 4
  for (int ks = 0; ks < 32; ++ks) {
    const int k = ks * 4 + 2 * hi;
    const float* pa = sA + (m0 + lo) * LDA + k;
    v2f a;
    a[0] = pa[0];
    a[1] = pa[1];
#pragma unroll
    for (int nt = 0; nt < 8; ++nt) {
      const float* pb = sB + (nt * 16 + lo) * LDB + k;
      v2f b;
      b[0] = pb[0];
      b[1] = pb[1];
      acc[nt] = __builtin_amdgcn_wmma_f32_16x16x4_f32(
          false, a, false, b, (short)0, acc[nt], false, false);
    }
  }
}

// ---- fused layer GEMM: out = relu( sum_r agg8[r] @ W[r] + x @ Wself + b ) --

__global__ __launch_bounds__(256) void gemm_fused(const float* __restrict__ x,
                                                  const float* __restrict__ agg8,
                                                  const float* __restrict__ W,
                                                  const float* __restrict__ Wself,
                                                  const float* __restrict__ bias,
                                                  float* __restrict__ out) {
  __shared__ float sA[128 * LDA];
  __shared__ float sB[128 * LDB];

  const int tid  = threadIdx.x;
  const int wave = tid >> 5;
  const int lane = tid & 31;
  const int lo   = lane & 15;
  const int hi   = lane >> 4;
  const int m_block = blockIdx.x * 128;
  const int m0 = wave * 16;

  v8f acc[8];
#pragma unroll
  for (int nt = 0; nt < 8; ++nt) {
    float c0 = bias[nt * 16 + lo];
#pragma unroll
    for (int v = 0; v < 8; ++v) acc[nt][v] = c0;
  }

  for (int r = 0; r < N_RELS + 1; ++r) {
    const float* Asrc = (r < N_RELS)
                            ? agg8 + ((size_t)r * N_NODES + m_block) * DIM
                            : x + (size_t)m_block * DIM;
    const float* Bsrc = (r < N_RELS) ? W + (size_t)r * DIM * DIM : Wself;
    __syncthreads();                       // previous phase done reading LDS
    load_tiles_lds(sA, sB, Asrc, Bsrc, tid);
    __syncthreads();
    kloop_wmma(sA, sB, acc, m0, lo, hi);
  }

#pragma unroll
  for (int nt = 0; nt < 8; ++nt) {
#pragma unroll
    for (int v = 0; v < 8; ++v) {
      out[(size_t)(m_block + m0 + v + 8 * hi) * DIM + nt * 16 + lo] =
          fmaxf(acc[nt][v], 0.0f);
    }
  }
}

// ---- fallback: single-relation GEMM (beta / bias / relu variants) ---------

template <int BETA, int BIAS, int RELU>
__global__ __launch_bounds__(256) void gemm128(const float* __restrict__ A,
                                               const float* __restrict__ B,
                                               const float* __restrict__ bias,
                                               float* __restrict__ C) {
  __shared__ float sA[128 * LDA];
  __shared__ float sB[128 * LDB];

  const int tid  = threadIdx.x;
  const int wave = tid >> 5;
  const int lane = tid & 31;
  const int lo   = lane & 15;
  const int hi   = lane >> 4;
  const int m_block = blockIdx.x * 128;
  const int m0 = wave * 16;

  load_tiles_lds(sA, sB, A + (size_t)m_block * DIM, B, tid);
  __syncthreads();

  v8f acc[8];
#pragma unroll
  for (int nt = 0; nt < 8; ++nt) {
#pragma unroll
    for (int v = 0; v < 8; ++v) {
      float c0 = 0.0f;
      if (BIAS) c0 = bias[nt * 16 + lo];
      if (BETA) c0 += C[(size_t)(m_block + m0 + v + 8 * hi) * DIM + nt * 16 + lo];
      acc[nt][v] = c0;
    }
  }

  kloop_wmma(sA, sB, acc, m0, lo, hi);

#pragma unroll
  for (int nt = 0; nt < 8; ++nt) {
#pragma unroll
    for (int v = 0; v < 8; ++v) {
      float val = acc[nt][v];
      if (RELU) val = fmaxf(val, 0.0f);
      C[(size_t)(m_block + m0 + v + 8 * hi) * DIM + nt * 16 + lo] = val;
    }
  }
}

// ---- edge scatter kernels -------------------------------------------------

// Single pass: agg8[etype[e]][dst[e]] += x[src[e]]  (one wave per edge)
__global__ __launch_bounds__(256) void scatter_all(const float* __restrict__ x,
                                                   const int* __restrict__ src,
                                                   const int* __restrict__ dst,
                                                   const int* __restrict__ et,
                                                   float* __restrict__ agg8) {
  const int e = blockIdx.x * 8 + (threadIdx.x >> 5);
  if (e >= N_EDGES) return;
  const int lane = threadIdx.x & 31;
  const float4 v = ((const float4*)(x + (size_t)src[e] * DIM))[lane];
  float* o = agg8 + ((size_t)et[e] * N_NODES + (size_t)dst[e]) * DIM + lane * 4;
  atomicAdd(o + 0, v.x);
  atomicAdd(o + 1, v.y);
  atomicAdd(o + 2, v.z);
  atomicAdd(o + 3, v.w);
}

// Per-relation pass (fallback): agg[dst] += x[src] for edges with etype==r
__global__ __launch_bounds__(256) void scatter_rel(const float* __restrict__ x,
                                                   const int* __restrict__ src,
                                                   const int* __restrict__ dst,
                                                   const int* __restrict__ et,
                                                   float* __restrict__ agg,
                                                   int r) {
  const int e = blockIdx.x * 8 + (threadIdx.x >> 5);
  if (e >= N_EDGES) return;
  if (et[e] != r) return;
  const int lane = threadIdx.x & 31;
  const float4 v = ((const float4*)(x + (size_t)src[e] * DIM))[lane];
  float* o = agg + (size_t)dst[e] * DIM + lane * 4;
  atomicAdd(o + 0, v.x);
  atomicAdd(o + 1, v.y);
  atomicAdd(o + 2, v.z);
  atomicAdd(o + 3, v.w);
}

// ---- pool + head ----------------------------------------------------------

__global__ __launch_bounds__(128) void pool_max(const float* __restrict__ h2,
                                                float* __restrict__ pooled) {
  const int g = blockIdx.x;
  const int c = threadIdx.x;
  const float* p = h2 + (size_t)g * 1024 * DIM + c;
  float m = -3.402823466e+38f;
#pragma unroll 8
  for (int i = 0; i < 1024; ++i) m = fmaxf(m, p[(size_t)i * DIM]);
  pooled[g * DIM + c] = m;
}

__global__ __launch_bounds__(128) void classify(const float* __restrict__ pooled,
                                                const float* __restrict__ Wc,
                                                const float* __restrict__ bc,
                                                float* __restrict__ out) {
  __shared__ float sp[DIM];
  const int g = blockIdx.x;
  sp[threadIdx.x] = pooled[g * DIM + threadIdx.x];
  __syncthreads();
  if (threadIdx.x < N_CLASSES) {
    float s = bc[threadIdx.x];
#pragma unroll 4
    for (int k = 0; k < DIM; ++k) s += sp[k] * Wc[k * N_CLASSES + threadIdx.x];
    out[g * N_CLASSES + threadIdx.x] = s;
  }
}

// ---- driver ---------------------------------------------------------------

extern "C" void kernel_launch(void* const* d_in, const int* in_sizes, int n_in,
                              void* d_out, int out_size, void* d_ws, size_t ws_size,
                              hipStream_t stream) {
  (void)in_sizes; (void)n_in; (void)out_size;

  const float* h       = (const float*)d_in[0];
  const float* W1      = (const float*)d_in[1];
  const float* W1_self = (const float*)d_in[2];
  const float* b1      = (const float*)d_in[3];
  const float* W2      = (const float*)d_in[4];
  const float* W2_self = (const float*)d_in[5];
  const float* b2      = (const float*)d_in[6];
  const float* Wc      = (const float*)d_in[7];
  const float* bc      = (const float*)d_in[8];
  const int*   src     = (const int*)d_in[9];
  const int*   dst     = (const int*)d_in[10];
  const int*   etype   = (const int*)d_in[11];
  // graph_ids (d_in[12]) unused: contiguous 1024-node blocks by construction.

  const size_t feat_elems = (size_t)N_NODES * DIM;          // 8.39M floats
  const size_t fused_need =
      (2 + N_RELS) * feat_elems * sizeof(float) + (size_t)N_GRAPHS * DIM * sizeof(float);

  const dim3 gemm_grid(N_NODES / 128);
  const dim3 gemm_blk(256);
  const dim3 sc_grid(N_EDGES / 8);
  const dim3 sc_blk(256);

  float* ws = (float*)d_ws;

  if (ws_size >= fused_need) {
    // ---------------- fused path ----------------
    float* h1     = ws;
    float* h2     = h1 + feat_elems;
    float* agg8   = h2 + feat_elems;                       // [8][N][128]
    float* pooled = agg8 + (size_t)N_RELS * feat_elems;
    const size_t agg8_bytes = (size_t)N_RELS * feat_elems * sizeof(float);

    // Layer 1
    hipMemsetAsync(agg8, 0, agg8_bytes, stream);
    scatter_all<<<sc_grid, sc_blk, 0, stream>>>(h, src, dst, etype, agg8);
    gemm_fused<<<gemm_grid, gemm_blk, 0, stream>>>(h, agg8, W1, W1_self, b1, h1);

    // Layer 2
    hipMemsetAsync(agg8, 0, agg8_bytes, stream);
    scatter_all<<<sc_grid, sc_blk, 0, stream>>>(h1, src, dst, etype, agg8);
    gemm_fused<<<gemm_grid, gemm_blk, 0, stream>>>(h1, agg8, W2, W2_self, b2, h2);

    pool_max<<<N_GRAPHS, DIM, 0, stream>>>(h2, pooled);
    classify<<<N_GRAPHS, DIM, 0, stream>>>(pooled, Wc, bc, (float*)d_out);
  } else {
    // ---------------- fallback path ----------------
    float* h1     = ws;
    float* h2     = h1 + feat_elems;
    float* agg    = h2 + feat_elems;
    float* pooled = agg + feat_elems;
    const size_t agg_bytes = feat_elems * sizeof(float);

    gemm128<0, 1, 0><<<gemm_grid, gemm_blk, 0, stream>>>(h, W1_self, b1, h1);
    for (int r = 0; r < N_RELS; ++r) {
      hipMemsetAsync(agg, 0, agg_bytes, stream);
      scatter_rel<<<sc_grid, sc_blk, 0, stream>>>(h, src, dst, etype, agg, r);
      const float* Wr = W1 + (size_t)r * DIM * DIM;
      if (r == N_RELS - 1)
        gemm128<1, 0, 1><<<gemm_grid, gemm_blk, 0, stream>>>(agg, Wr, nullptr, h1);
      else
        gemm128<1, 0, 0><<<gemm_grid, gemm_blk, 0, stream>>>(agg, Wr, nullptr, h1);
    }

    gemm128<0, 1, 0><<<gemm_grid, gemm_blk, 0, stream>>>(h1, W2_self, b2, h2);
    for (int r = 0; r < N_RELS; ++r) {
      hipMemsetAsync(agg, 0, agg_bytes, stream);
      scatter_rel<<<sc_grid, sc_blk, 0, stream>>>(h1, src, dst, etype, agg, r);
      const float* Wr = W2 + (size_t)r * DIM * DIM;
      if (r == N_RELS - 1)
        gemm128<1, 0, 1><<<gemm_grid, gemm_blk, 0, stream>>>(agg, Wr, nullptr, h2);
      else
        gemm128<1, 0, 0><<<gemm_grid, gemm_blk, 0, stream>>>(agg, Wr, nullptr, h2);
    }

    pool_max<<<N_GRAPHS, DIM, 0, stream>>>(h2, pooled);
    classify<<<N_GRAPHS, DIM, 0, stream>>>(pooled, Wc, bc, (float*)d_out);
  }
}